// Qwen2VLInterleaveEmbeddings_13134009991215
// MI455X (gfx1250) — compile-verified
//
#include <hip/hip_runtime.h>

// Qwen2VL interleave-embeddings: pure bandwidth-bound scatter/copy.
// B=4, S=4096, D=2048, NV=1024, fp32. Roofline: ~256 MiB @ 23.3 TB/s ~= 11 us.
// No FLOPs -> no WMMA; optimize for b128 VMEM with non-temporal hints.

typedef float f4 __attribute__((ext_vector_type(4)));

constexpr int B      = 4;
constexpr int S      = 4096;
constexpr int D      = 2048;
constexpr int NV     = 1024;
constexpr int D4     = D / 4;      // 512 float4 per row
constexpr int NROWS  = B * S;      // 16384 output rows
constexpr int NVROWS = B * NV;     // 4096 vision rows

// ---------------- map-based single-pass path (uses workspace) ----------------

__global__ void init_map_kernel(int* __restrict__ map) {
    int i = blockIdx.x * blockDim.x + threadIdx.x;
    if (i < NROWS) map[i] = -1;
}

__global__ void scatter_map_kernel(const int* __restrict__ vidx,
                                   int* __restrict__ map) {
    int i = blockIdx.x * blockDim.x + threadIdx.x;
    if (i < NVROWS) {
        int b = i >> 10;               // i / NV
        int s = vidx[i];
        // s == 0 excluded: reference restores original text row 0 per batch,
        // so any vision write at position 0 must lose to text.
        if (s > 0 && s < S) map[b * S + s] = i;
    }
}

__global__ void __launch_bounds__(D4)
interleave_kernel(const f4* __restrict__ vis,
                  const f4* __restrict__ txt,
                  const int* __restrict__ map,
                  f4* __restrict__ out) {
    const int row = blockIdx.x;        // 0 .. NROWS-1 (uniform per block)
    const int col = threadIdx.x;       // 0 .. D4-1
    const int m   = map[row];          // block-uniform -> scalar load
    const f4* src = (m >= 0) ? (vis + (size_t)m * D4)
                             : (txt + (size_t)row * D4);
    f4 v = __builtin_nontemporal_load(src + col);       // global_load_b128 NT
    __builtin_nontemporal_store(v, out + (size_t)row * D4 + col); // b128 NT
}

// ---------------- fallback path (no workspace): copy then scatter -----------

__global__ void __launch_bounds__(512)
copy_kernel(const f4* __restrict__ txt, f4* __restrict__ out) {
    size_t i = (size_t)blockIdx.x * blockDim.x + threadIdx.x;
    f4 v = __builtin_nontemporal_load(txt + i);
    __builtin_nontemporal_store(v, out + i);
}

__global__ void __launch_bounds__(D4)
scatter_rows_kernel(const f4* __restrict__ vis,
                    const int* __restrict__ vidx,
                    f4* __restrict__ out) {
    const int i   = blockIdx.x;        // vision row (uniform per block)
    const int col = threadIdx.x;
    const int s   = vidx[i];           // block-uniform -> scalar load
    if (s > 0 && s < S) {              // skip s==0: text row 0 is restored
        const int b = i >> 10;
        f4 v = __builtin_nontemporal_load(vis + (size_t)i * D4 + col);
        __builtin_nontemporal_store(v, out + ((size_t)b * S + s) * D4 + col);
    }
}

// ----------------------------------------------------------------------------

extern "C" void kernel_launch(void* const* d_in, const int* in_sizes, int n_in,
                              void* d_out, int out_size, void* d_ws, size_t ws_size,
                              hipStream_t stream) {
    const f4*  vis  = (const f4*)d_in[0];   // [B*NV, D] float32
    const f4*  txt  = (const f4*)d_in[1];   // [B, S, D] float32
    const int* vidx = (const int*)d_in[2];  // [B, NV] int32
    f4*        out  = (f4*)d_out;           // [B, S, D] float32

    if (d_ws && ws_size >= (size_t)NROWS * sizeof(int)) {
        // Single-pass: 128 MiB read + 128 MiB write (+64 KB map traffic).
        int* map = (int*)d_ws;
        init_map_kernel<<<(NROWS + 255) / 256, 256, 0, stream>>>(map);
        scatter_map_kernel<<<(NVROWS + 255) / 256, 256, 0, stream>>>(vidx, map);
        interleave_kernel<<<NROWS, D4, 0, stream>>>(vis, txt, map, out);
    } else {
        // Fallback two-pass: 160 MiB read + 160 MiB write.
        copy_kernel<<<(NROWS * D4) / 512, 512, 0, stream>>>(txt, out);
        scatter_rows_kernel<<<NVROWS, D4, 0, stream>>>(vis, vidx, out);
    }
}